// GuidedFilter_88819923681600
// MI455X (gfx1250) — compile-verified
//
#include <hip/hip_runtime.h>

// Guided filter for MI455X (gfx1250), two fused tiled stages.
// Stage 1: guide,input -> a,b (written to d_ws).
// Stage 2: a,b,guide  -> out.
// Box filters done per 32x32 tile in LDS: vertical pass = scalar sliding sums,
// horizontal pass = fp32 WMMA against a 0/1 band matrix (exact fp32 math).
// Interior tiles fill LDS with the Tensor Data Mover (tensor_load_to_lds);
// border tiles use a scalar reflect-padding loop.

typedef float v2f __attribute__((ext_vector_type(2)));
typedef float v8f __attribute__((ext_vector_type(8)));
typedef unsigned int v4u __attribute__((ext_vector_type(4)));
typedef int v4i __attribute__((ext_vector_type(4)));
typedef int v8i __attribute__((ext_vector_type(8)));

#define DEV __device__ __forceinline__

constexpr int Hs = 1024, Ws = 1024, Bs = 16;
constexpr int R = 15;                 // filter radius
constexpr int KW = 2 * R + 1;         // 31-tap window
constexpr int TILE = 32;              // output tile per workgroup
constexpr int HALO = TILE + 2 * R;    // 62 = input rows per tile
constexpr int RAW_STRIDE = 64;        // LDS row stride for raw halo (TDM-contiguous)
constexpr int VS_STRIDE = 65;         // LDS row stride for vertical sums (bank-friendly)
constexpr float INV_KK = 1.0f / (31.0f * 31.0f);
constexpr float EPSV = 0.5f;

DEV int mirror(int i, int n) {        // jnp.pad mode='reflect' (no edge repeat)
  i = (i < 0) ? -i : i;
  return (i >= n) ? (2 * n - 2 - i) : i;
}

// Build TDM descriptor group1 for a 2D fp32 tile copy: tile_dim0 x tile_dim1
// dwords, global row stride `stride0` dwords, huge tensor dims (no OOB).
DEV v8i tdm_group1_2d(unsigned tile_d0, unsigned tile_d1, unsigned stride0) {
  const unsigned td_big = 1u << 20;  // tensor_dim0/1: anything >= tile dims
  v8i g1;
  g1[0] = (int)(2u << 16);  // workgroup_mask=0 | data_size=2 (4B) | no flags
  g1[1] = (int)((td_big & 0xFFFFu) << 16);                         // [63:48] td0 lo
  g1[2] = (int)(((td_big >> 16) & 0xFFFFu) | ((td_big & 0xFFFFu) << 16));  // td0 hi | td1 lo
  g1[3] = (int)(((td_big >> 16) & 0xFFFFu) | (tile_d0 << 16));     // td1 hi | tile_dim0
  g1[4] = (int)(tile_d1 & 0xFFFFu);                                // tile_dim1 | tile_dim2=0
  g1[5] = (int)stride0;                                            // tensor_dim0_stride lo32
  g1[6] = 0;                                                       // stride0 hi16 | stride1 lo16
  g1[7] = 0;                                                       // stride1 hi32
  return g1;
}

DEV v4u tdm_group0(unsigned lds_addr, unsigned long long gaddr) {
  v4u g0;
  g0[0] = 1u;                                // count=1 (valid descriptor)
  g0[1] = lds_addr;                          // LDS byte address
  g0[2] = (unsigned)gaddr;                   // global_addr[31:0]
  g0[3] = ((unsigned)(gaddr >> 32) & 0x01FFFFFFu) | (2u << 30);  // addr[56:32] | type=2
  return g0;
}

// NCH: channels to box-filter. MODE 0: stats stage (4ch). MODE 1: output stage (2ch).
template <int NCH, int MODE>
__global__ __launch_bounds__(256) void gf_stage(const float* __restrict__ in0,
                                                const float* __restrict__ in1,
                                                const float* __restrict__ guide,
                                                float* __restrict__ out0,
                                                float* __restrict__ out1) {
  __shared__ float raw[2][HALO][RAW_STRIDE];   // 31,744 B
  __shared__ float vs[NCH][TILE][VS_STRIDE];   // 33,280 B (NCH=4) / 16,640 B (NCH=2)
  float* meanbuf = &raw[0][0][0];              // aliases raw after it is dead
                                               // needs NCH*TILE*TILE = 4096 floats, raw has 7936

  const int tid = threadIdx.x;
  const int img = blockIdx.z;
  const int y0 = blockIdx.y * TILE;
  const int x0 = blockIdx.x * TILE;
  const size_t base = (size_t)img * ((size_t)Hs * Ws);

  // Warm caches for the epilogue's guide reads (emits global_prefetch_b8).
  if (MODE == 1) {
    for (int p = tid; p < TILE * TILE; p += 256) {
      int r = p >> 5, c = p & 31;
      __builtin_prefetch(&guide[base + (size_t)(y0 + r) * Ws + (x0 + c)], 0, 0);
    }
  }

  // ---- Phase A: fill raw[0]=in0 halo, raw[1]=in1 halo ----
  const float* p0 = in0 + base;
  const float* p1 = in1 + base;
  // Interior tile: rows y0-R..y0-R+61 and 64 cols x0-R..x0-R+63 all in-image.
  const bool interior = (y0 - R >= 0) && (y0 - R + HALO <= Hs) &&
                        (x0 - R >= 0) && (x0 - R + RAW_STRIDE <= Ws);
  if (interior) {
    // One TDM DMA per buffer: 62 rows x 64 dwords, global row stride 1024.
    if (__builtin_amdgcn_readfirstlane(tid) == 0) {  // wave 0 issues (EXEC ignored by TDM)
      const v8i g1 = tdm_group1_2d(RAW_STRIDE, HALO, Ws);
      const v4i z4 = {0, 0, 0, 0};
      const v8i z8 = {0, 0, 0, 0, 0, 0, 0, 0};
      const size_t goff = (size_t)(y0 - R) * Ws + (x0 - R);
      unsigned long long ga0 = (unsigned long long)(uintptr_t)(p0 + goff);
      unsigned long long ga1 = (unsigned long long)(uintptr_t)(p1 + goff);
      unsigned lds0 = (unsigned)(uintptr_t)&raw[0][0][0];
      unsigned lds1 = (unsigned)(uintptr_t)&raw[1][0][0];
      __builtin_amdgcn_tensor_load_to_lds(tdm_group0(lds0, ga0), g1, z4, z4, z8, 0);
      __builtin_amdgcn_tensor_load_to_lds(tdm_group0(lds1, ga1), g1, z4, z4, z8, 0);
      __builtin_amdgcn_s_wait_tensorcnt(0);
    }
  } else {
    // Border tile: scalar gather with reflect mirroring.
    for (int idx = tid; idx < HALO * HALO; idx += 256) {
      int rr = idx / HALO, cc = idx % HALO;
      int gy = mirror(y0 - R + rr, Hs);
      int gx = mirror(x0 - R + cc, Ws);
      size_t g = (size_t)gy * Ws + gx;
      raw[0][rr][cc] = p0[g];
      raw[1][rr][cc] = p1[g];
    }
  }
  __syncthreads();

  // ---- Phase B: vertical 31-tap sliding sums (one column per thread-task) ----
  // vs[ch][r][c] = sum_{k=0..30} val(ch, r+k, c), for r in [0,32), c in [0,64).
  for (int t = tid; t < NCH * 64; t += 256) {
    const int ch = t >> 6;
    const int c = t & 63;
    if (c >= HALO) {
      // Zero-fill pad columns: WMMA lanes with weight 0 must not see garbage
      // (0 * NaN -> NaN per ISA 7.12).
      for (int r = 0; r < TILE; ++r) vs[ch][r][c] = 0.0f;
    } else {
      auto val = [&](int r) -> float {
        if (MODE == 0) {
          float a0 = raw[0][r][c], a1 = raw[1][r][c];
          return (ch == 0) ? a0 : (ch == 1) ? a1 : (ch == 2) ? a0 * a1 : a0 * a0;
        } else {
          return raw[ch][r][c];
        }
      };
      float s = 0.0f;
      for (int r = 0; r < KW; ++r) s += val(r);
      vs[ch][0][c] = s;
      for (int r = 1; r < TILE; ++r) {
        s += val(r + KW - 1) - val(r - 1);
        vs[ch][r][c] = s;
      }
    }
  }
  __syncthreads();

  // ---- Phase C: horizontal 31-tap sum as WMMA f32 16x16x4 chain ----
  // Out[r][n] = sum_{j=n..n+30} vs[r][ct*16 + j]  == A(16xK) x Band(Kx16), K=48.
  // A frag per ISA: VGPR v, lane l -> M = l&15, K = 4*step + v + 2*(l>>4).
  // Band frag uses the same per-lane K mapping (sum is invariant to a shared K
  // permutation), N = l&15.
  {
    const int wave = tid >> 5;
    const int lane = tid & 31;
    const int lhalf = lane >> 4;
    const int lrow = lane & 15;
    constexpr int NPOS = (TILE / 16) * (TILE / 16);  // 4 16x16 tiles per block
    for (int job = wave; job < NPOS * NCH; job += 8) {
      const int ch = job / NPOS;
      const int pos = job % NPOS;
      const int rt = pos >> 1, ct = pos & 1;
      v8f acc = {};
      for (int step = 0; step < 12; ++step) {
        const int k0 = 4 * step + 2 * lhalf;  // krel of this lane's VGPR0 slot
        v2f a, b;
        a.x = vs[ch][rt * 16 + lrow][ct * 16 + k0];
        a.y = vs[ch][rt * 16 + lrow][ct * 16 + k0 + 1];
        const int n = lrow;
        b.x = (k0 >= n && k0 <= n + 2 * R) ? 1.0f : 0.0f;
        b.y = (k0 + 1 >= n && k0 + 1 <= n + 2 * R) ? 1.0f : 0.0f;
        acc = __builtin_amdgcn_wmma_f32_16x16x4_f32(false, a, false, b,
                                                    (short)0, acc, false, false);
      }
      // C/D layout: VGPR v -> row = v + 8*(l>>4), col = l&15.
#pragma unroll
      for (int v = 0; v < 8; ++v) {
        const int r = rt * 16 + v + 8 * lhalf;
        const int c = ct * 16 + lrow;
        meanbuf[(ch * TILE + r) * TILE + c] = acc[v] * INV_KK;
      }
    }
  }
  __syncthreads();

  // ---- Phase D: elementwise epilogue ----
  const size_t obase = base + (size_t)y0 * Ws + x0;
  for (int p = tid; p < TILE * TILE; p += 256) {
    const int r = p >> 5, c = p & 31;
    const size_t g = obase + (size_t)r * Ws + c;
    if (MODE == 0) {
      float mI = meanbuf[(0 * TILE + r) * TILE + c];
      float mp = meanbuf[(1 * TILE + r) * TILE + c];
      float mIp = meanbuf[(2 * TILE + r) * TILE + c];
      float mII = meanbuf[(3 * TILE + r) * TILE + c];
      float cov = mIp - mI * mp;
      float var = mII - mI * mI;
      float aa = cov / (var + EPSV);
      float bb = mp - aa * mI;
      out0[g] = aa;
      out1[g] = bb;
    } else {
      float ma = meanbuf[(0 * TILE + r) * TILE + c];
      float mb = meanbuf[(1 * TILE + r) * TILE + c];
      out0[g] = ma * guide[g] + mb;
    }
  }
}

extern "C" void kernel_launch(void* const* d_in, const int* in_sizes, int n_in,
                              void* d_out, int out_size, void* d_ws, size_t ws_size,
                              hipStream_t stream) {
  const float* guide = (const float*)d_in[0];
  const float* input_map = (const float*)d_in[1];
  float* out = (float*)d_out;

  const size_t N = (size_t)Bs * Hs * Ws;  // 16M px
  // Workspace: a then b, 2*N floats = 128 MB.
  float* wa = (float*)d_ws;
  float* wb = wa + N;

  dim3 grid(Ws / TILE, Hs / TILE, Bs);  // (32, 32, 16)
  dim3 block(256);

  gf_stage<4, 0><<<grid, block, 0, stream>>>(guide, input_map, nullptr, wa, wb);
  gf_stage<2, 1><<<grid, block, 0, stream>>>(wa, wb, guide, out, nullptr);
}